// TimeAwareMultiHeadAttention_3384434229848
// MI455X (gfx1250) — compile-verified
//
#include <hip/hip_runtime.h>
#include <hip/hip_bf16.h>
#include <cmath>

// Problem dims (fixed by the reference)
constexpr int Bn = 8, Ln = 256, Dn = 128, Hn = 4, HSn = 32;
#define NEGV (-4294967295.0f)

typedef __attribute__((ext_vector_type(2))) float v2f;
typedef __attribute__((ext_vector_type(4))) float f4;
typedef __attribute__((ext_vector_type(8))) float v8f;

// =====================================================================
// Kernel 1: fp32 WMMA projections.
//   gemm 0: Qs   = (queries @ Wq^T + bq) * (1/sqrt(HS))   (scale folded in)
//   gemm 1: Keff =  keys    @ Wk^T + bk + abs_pos_K       (QK + QpK fusion)
//   gemm 2: Veff =  keys    @ Wv^T + bv + abs_pos_V       (AV + ApV fusion)
// M = B*L = 2048, N = 128, K = 128. One wave per 16x16 tile,
// V_WMMA_F32_16X16X4_F32 stepping K by 4 (32 wmma ops per tile).
// =====================================================================
__global__ __launch_bounds__(32)
void proj_kernel(const float* __restrict__ queries,
                 const float* __restrict__ keys,
                 const float* __restrict__ Wq, const float* __restrict__ bq,
                 const float* __restrict__ Wk, const float* __restrict__ bk,
                 const float* __restrict__ Wv, const float* __restrict__ bv,
                 const float* __restrict__ posK, const float* __restrict__ posV,
                 float* __restrict__ Qs, float* __restrict__ Keff,
                 float* __restrict__ Veff)
{
    const int lane = threadIdx.x;      // wave32
    const int half = lane >> 4;        // 0: lanes 0-15, 1: lanes 16-31
    const int lm   = lane & 15;

    constexpr int tiles_per_gemm = (Bn * Ln / 16) * (Dn / 16);  // 128*8 = 1024
    const int gemm = blockIdx.x / tiles_per_gemm;
    const int t    = blockIdx.x % tiles_per_gemm;
    const int n0   = (t >> 3) * 16;    // row tile base in [0, 2048)
    const int j0   = (t & 7) * 16;     // col tile base in [0, 128)

    const float* in;  const float* W;  const float* bias;
    const float* extra; float* out; float scale;
    if (gemm == 0)      { in = queries; W = Wq; bias = bq; extra = nullptr; out = Qs;   scale = 0.17677669529663687f; }
    else if (gemm == 1) { in = keys;    W = Wk; bias = bk; extra = posK;    out = Keff; scale = 1.0f; }
    else                { in = keys;    W = Wv; bias = bv; extra = posV;    out = Veff; scale = 1.0f; }

    // Per-lane source rows. A[m][k] = in[n0+m][k]; B[k][j] = W[j0+j][k] (W is [j][k], contraction over k).
    const float* arow = in + (size_t)(n0 + lm) * Dn;
    const float* brow = W  + (size_t)(j0 + lm) * Dn;

#if __has_builtin(__builtin_amdgcn_wmma_f32_16x16x4_f32)
    v8f acc = {0.f, 0.f, 0.f, 0.f, 0.f, 0.f, 0.f, 0.f};
    for (int k0 = 0; k0 < Dn; k0 += 4) {
        // ISA 16x4 fp32 A layout: VGPR0 = {K=k0 | K=k0+2}, VGPR1 = {K=k0+1 | K=k0+3}
        // across the two half-waves; B (4x16) mirrors this with rows across halves.
        const int ko = k0 + 2 * half;
        v2f a; a[0] = arow[ko]; a[1] = arow[ko + 1];
        v2f b; b[0] = brow[ko]; b[1] = brow[ko + 1];
        acc = __builtin_amdgcn_wmma_f32_16x16x4_f32(
            /*neg_a=*/false, a, /*neg_b=*/false, b,
            /*c_mod=*/(short)0, acc, /*reuse_a=*/false, /*reuse_b=*/false);
    }
#else
    // Scalar fallback (keeps compile green if builtin name differs; histogram
    // will show wmma==0 and we iterate).
    float accf[8] = {0.f, 0.f, 0.f, 0.f, 0.f, 0.f, 0.f, 0.f};
    for (int k = 0; k < Dn; ++k) {
        const float bvv = brow[k];
        #pragma unroll
        for (int r = 0; r < 8; ++r)
            accf[r] += in[(size_t)(n0 + r + 8 * half) * Dn + k] * bvv;
    }
    v8f acc;
    #pragma unroll
    for (int r = 0; r < 8; ++r) acc[r] = accf[r];
#endif

    // D layout: VGPR r -> M = r (lanes 0-15) / r+8 (lanes 16-31), N = lane%16.
    const int j = j0 + lm;
    const float bj = bias[j];
    #pragma unroll
    for (int r = 0; r < 8; ++r) {
        const int m = n0 + r + 8 * half;
        float v = acc[r] + bj;
        if (extra) v += extra[(size_t)m * Dn + j];
        out[(size_t)m * Dn + j] = v * scale;
    }
}

// =====================================================================
// Kernel 2: fused time-aware attention. One 256-thread block per (b,q).
//   attn[h][k] = Qs[b,q,head h] . (Keff[b,k] + tK[b,q,k])    (Qs pre-scaled)
//   softmax over k with row/causal mask
//   out[b,q,d] = sum_k A[h(d)][k] * (Veff[b,k,d] + tV[b,q,k,d])
// tK/tV rows streamed once, non-temporal b128 loads, wave-per-32-k-rows.
// =====================================================================
__global__ __launch_bounds__(256)
void attn_kernel(const float* __restrict__ Qs,
                 const float* __restrict__ Keff,
                 const float* __restrict__ Veff,
                 const float* __restrict__ tK,
                 const float* __restrict__ tV,
                 const unsigned char* __restrict__ time_mask,  // numpy bool layout
                 float* __restrict__ out)
{
    __shared__ __align__(16) float qsh[Dn];
    __shared__ float attnS[Hn][Ln];
    __shared__ __align__(16) float partial[8][Dn];

    const int tid  = threadIdx.x;
    const int wave = tid >> 5;
    const int lane = tid & 31;
    const int bq   = blockIdx.x;       // = b*L + q
    const int b    = bq >> 8;
    const int q    = bq & 255;

    if (tid < Dn) qsh[tid] = Qs[(size_t)bq * Dn + tid];
    __syncthreads();

    const int  head    = lane >> 3;                 // lane covers d = 4*lane..4*lane+3
    const f4   q4      = *(const f4*)(qsh + 4 * lane);
    const bool rowmask = (time_mask[bq] != 0);

    const float* tKb = tK   + (size_t)bq * Ln * Dn;
    const float* Keb = Keff + (size_t)b  * Ln * Dn;

    // ---- Phase 1: attention logits (wave w owns k in [32w, 32w+32)) ----
    for (int i = 0; i < 32; ++i) {
        const int k = (wave << 5) + i;
        const float* tkp = tKb + (size_t)k * Dn + 4 * lane;
        if (i + 4 < 32) __builtin_prefetch(tkp + 4 * Dn, 0, 1);   // global_prefetch_b8
        f4 tk = __builtin_nontemporal_load((const f4*)tkp);       // streamed once -> NT
        f4 ke = *(const f4*)(Keb + (size_t)k * Dn + 4 * lane);    // L2-resident
        float s = q4.x * (tk.x + ke.x) + q4.y * (tk.y + ke.y)
                + q4.z * (tk.z + ke.z) + q4.w * (tk.w + ke.w);
        // reduce 8 lanes (one head = 32 d's = 8 lanes of f4)
        s += __shfl_xor(s, 1, 32);
        s += __shfl_xor(s, 2, 32);
        s += __shfl_xor(s, 4, 32);
        if ((lane & 7) == 0)
            attnS[head][k] = (rowmask || (k > q)) ? NEGV : s;
    }
    __syncthreads();

    // ---- Phase 2: per-head softmax over 256 keys (wave h handles head h) ----
    if (wave < Hn) {
        float vals[8];
        float m = -3.4e38f;
        #pragma unroll
        for (int j = 0; j < 8; ++j) {
            vals[j] = attnS[wave][lane + 32 * j];
            m = fmaxf(m, vals[j]);
        }
        for (int off = 16; off > 0; off >>= 1) m = fmaxf(m, __shfl_xor(m, off, 32));
        float sum = 0.f;
        #pragma unroll
        for (int j = 0; j < 8; ++j) {
            const float e = __expf(vals[j] - m);
            vals[j] = e; sum += e;
        }
        for (int off = 16; off > 0; off >>= 1) sum += __shfl_xor(sum, off, 32);
        const float inv = 1.0f / sum;
        #pragma unroll
        for (int j = 0; j < 8; ++j) attnS[wave][lane + 32 * j] = vals[j] * inv;
    }
    __syncthreads();

    // ---- Phase 3: weighted value accumulation ----
    const float* tVb = tV   + (size_t)bq * Ln * Dn;
    const float* Veb = Veff + (size_t)b  * Ln * Dn;
    f4 acc = {0.f, 0.f, 0.f, 0.f};
    for (int i = 0; i < 32; ++i) {
        const int k = (wave << 5) + i;
        const float a = attnS[head][k];
        const float* tvp = tVb + (size_t)k * Dn + 4 * lane;
        if (i + 4 < 32) __builtin_prefetch(tvp + 4 * Dn, 0, 1);
        f4 tv = __builtin_nontemporal_load((const f4*)tvp);
        f4 ve = *(const f4*)(Veb + (size_t)k * Dn + 4 * lane);
        acc += a * (tv + ve);
    }
    *(f4*)(&partial[wave][4 * lane]) = acc;
    __syncthreads();

    if (tid < Dn) {
        float r = 0.f;
        #pragma unroll
        for (int w = 0; w < 8; ++w) r += partial[w][tid];
        out[(size_t)bq * Dn + tid] = r;
    }
}

// =====================================================================
// Launch
// =====================================================================
extern "C" void kernel_launch(void* const* d_in, const int* in_sizes, int n_in,
                              void* d_out, int out_size, void* d_ws, size_t ws_size,
                              hipStream_t stream) {
    const float* queries = (const float*)d_in[0];
    const float* keys    = (const float*)d_in[1];
    const unsigned char* time_mask = (const unsigned char*)d_in[2];  // bool[B,L]
    // d_in[3] attn_mask is strict-causal (triu, k=1): computed inline as (k > q).
    const float* tK   = (const float*)d_in[4];
    const float* tV   = (const float*)d_in[5];
    const float* posK = (const float*)d_in[6];
    const float* posV = (const float*)d_in[7];
    const float* Wq = (const float*)d_in[8];
    const float* bq = (const float*)d_in[9];
    const float* Wk = (const float*)d_in[10];
    const float* bk = (const float*)d_in[11];
    const float* Wv = (const float*)d_in[12];
    const float* bv = (const float*)d_in[13];
    float* out = (float*)d_out;

    // Workspace: Qs | Keff | Veff, each B*L*D f32 (1 MB each, 3 MB total).
    float* Qsw  = (float*)d_ws;
    float* Keff = Qsw  + (size_t)Bn * Ln * Dn;
    float* Veff = Keff + (size_t)Bn * Ln * Dn;

    constexpr int tiles_per_gemm = (Bn * Ln / 16) * (Dn / 16);  // 1024
    proj_kernel<<<3 * tiles_per_gemm, 32, 0, stream>>>(
        queries, keys, Wq, bq, Wk, bk, Wv, bv, posK, posV, Qsw, Keff, Veff);

    attn_kernel<<<Bn * Ln, 256, 0, stream>>>(
        Qsw, Keff, Veff, tK, tV, time_mask, out);
}